// StackedBinaryWeight3D_4045859193374
// MI455X (gfx1250) — compile-verified
//
#include <hip/hip_runtime.h>

// out[( (k*256 + c)*256 + r )*27 + j] = (j==k) ? sigmoid(loc[(k*256 + r)*256 + c]) : 0
// K=27, DIN=DOUT=256. Output = 47,775,744 f32 (191 MB): pure store-bandwidth problem
// (roofline ~8.5 us at 23.3 TB/s; no matmul -> no WMMA role). One workgroup per output
// row (k,c): k is scalar, so per-thread diagonal classification is loop-invariant.
// Hot loop: value float4s = ds_load_b128 of a prebuilt LDS pattern + nontemporal
// global_store_b128; zero float4s = gfx1250 async LDS->global b128 store (ASYNCcnt).

namespace {

typedef float v4f __attribute__((ext_vector_type(4)));

constexpr unsigned kK = 27u;
// per row: 256 segments * 27 floats = 6912 floats = 1728 float4s = 8 iters * 216 threads

__global__ __launch_bounds__(216) void sbw3d_row_kernel(const float* __restrict__ loc,
                                                        float* __restrict__ out) {
  __shared__ v4f   patt[256];  // full 16B content of the float4 holding diagonal r
  __shared__ float zq[4];      // 16B of zeros: source for async LDS->global stores

  const unsigned g = blockIdx.x;   // output row = k*256 + c, 0..6911
  const unsigned k = g >> 8;       // 0..26 (scalar)
  const unsigned c = g & 255u;     // 0..255 (scalar)
  const unsigned t = threadIdx.x;  // 0..215

  // Phase 1: build diagonal float4 patterns. Diagonal r lives at float pos r*27+k
  // within the row => component (27r+k)&3 = (3r+k)&3 of its float4. The owning
  // thread builds the whole 16B pattern, so a single __syncthreads suffices.
  for (unsigned r = t; r < 256u; r += 216u) {
    const float x = loc[(k << 16) | (r << 8) | c];             // loc[k, r, c]
    const float s = __builtin_amdgcn_rcpf(1.0f + __expf(-x));  // v_rcp + v_exp
    const unsigned comp = (3u * r + k) & 3u;
    const v4f p = { comp == 0u ? s : 0.0f, comp == 1u ? s : 0.0f,
                    comp == 2u ? s : 0.0f, comp == 3u ? s : 0.0f };
    patt[r] = p;
  }
  if (t < 4u) zq[t] = 0.0f;
  __syncthreads();

  // Setup (only divmods in the kernel): thread t always covers local float4 index
  // l = t%27 of a 4-segment group; group index = t/27 + 8*it. Diagonal of segment i
  // in a group sits in local float4 (i*27+k)>>2 -- constant per row.
  const unsigned l  = t % kK;
  const unsigned g0 = t / kK;  // 0..7
  int spec_i = -1;
#pragma unroll
  for (unsigned i = 0u; i < 4u; ++i) {
    if (((i * kK + k) >> 2) == l) spec_i = (int)i;
  }

  const unsigned base4 = g * 1728u + t;  // this thread's first float4 index
  v4f* const out4 = reinterpret_cast<v4f*>(out);

  if (spec_i >= 0) {
    // 4/27 of threads: each float4 carries diagonal r = 4*(g0+8*it) + spec_i.
    const unsigned r0 = 4u * g0 + (unsigned)spec_i;
#pragma unroll
    for (unsigned it = 0u; it < 8u; ++it) {
      const v4f v = patt[r0 + 32u * it];                     // ds_load_b128
      __builtin_nontemporal_store(v, out4 + base4 + 216u * it);  // b128 NT store
    }
  } else {
    // 23/27 of threads: pure-zero float4s via gfx1250 async store, LDS -> global,
    // 16B per lane. mem_addr = SGPR base + u32 VGPR offset (191MB fits u32).
    // ASYNCcnt-tracked; S_ENDPGM's implicit wait-idle drains it. Bytes are disjoint
    // from the NT-store path, so no ordering is required.
    const unsigned lds_zero = (unsigned)(size_t)&zq[0];
    const unsigned long long gbase = (unsigned long long)(size_t)out;
#pragma unroll
    for (unsigned it = 0u; it < 8u; ++it) {
      const unsigned voff = (base4 + 216u * it) * 16u;
      asm volatile("global_store_async_from_lds_b128 %0, %1, %2"
                   :
                   : "v"(voff), "v"(lds_zero), "s"(gbase)
                   : "memory");
    }
  }
}

}  // namespace

extern "C" void kernel_launch(void* const* d_in, const int* in_sizes, int n_in,
                              void* d_out, int out_size, void* d_ws, size_t ws_size,
                              hipStream_t stream) {
  (void)in_sizes; (void)n_in; (void)d_ws; (void)ws_size; (void)out_size;

  const float* loc = (const float*)d_in[0];
  float* out = (float*)d_out;

  // 27*256 = 6912 output rows, one workgroup each; 216 threads = 7 wave32 waves.
  sbw3d_row_kernel<<<27u * 256u, 216u, 0, stream>>>(loc, out);
}